// MultiscaleVectorQuantizer2_81071802679601
// MI455X (gfx1250) — compile-verified
//
#include <hip/hip_runtime.h>

typedef __attribute__((ext_vector_type(2))) float v2f;
typedef __attribute__((ext_vector_type(8))) float v8f;

#define B_      32
#define C_      64
#define H_      32
#define V_      4096
#define FHAT_N  (B_*C_*H_*H_)   // 2097152

// ---------------- reduction helper ----------------
__device__ __forceinline__ float blockReduceSum256(float v, float* sdata) {
    int tid = threadIdx.x;
    sdata[tid] = v; __syncthreads();
    for (int s = 128; s > 0; s >>= 1) {
        if (tid < s) sdata[tid] += sdata[tid + s];
        __syncthreads();
    }
    return sdata[0];
}

// ---------------- init: f_rest = f, f_hat = 0, commit_acc = 0 ----------------
__global__ void init_kernel(const float* __restrict__ f, float* __restrict__ f_rest,
                            float* __restrict__ f_hat, float* __restrict__ commit_acc) {
    int i = blockIdx.x * blockDim.x + threadIdx.x;
    if (i < FHAT_N) { f_rest[i] = f[i]; f_hat[i] = 0.0f; }
    if (i == 0) *commit_acc = 0.0f;
}

// ---------------- e_sq[v] = ||E[v]||^2 ----------------
__global__ void esq_kernel(const float* __restrict__ E, float* __restrict__ esq) {
    int v = blockIdx.x * blockDim.x + threadIdx.x;
    if (v >= V_) return;
    const float* row = E + v * 64;
    float s = 0.0f;
    #pragma unroll
    for (int c = 0; c < 64; ++c) s += row[c] * row[c];
    esq[v] = s;
}

// ---------------- wmod[k][o][i][*] = conv_w * (1 + fc_w*6 + fc_b) ----------------
__global__ void wmod_kernel(const float* __restrict__ cw, const float* __restrict__ fcw,
                            const float* __restrict__ fcb, float* __restrict__ wmod) {
    int t = blockIdx.x * blockDim.x + threadIdx.x;   // t = k*4096 + (o*64+i)
    if (t >= 4 * 64 * 64) return;
    float scale = 1.0f + fcw[t] * 6.0f + fcb[t];
    #pragma unroll
    for (int j = 0; j < 9; ++j) wmod[t * 9 + j] = cw[t * 9 + j] * scale;
}

// ---------------- downsample residual into row-major [N,64] (exact block mean) ----------------
__global__ void down_nc_kernel(const float* __restrict__ f_rest, float* __restrict__ X, int pn) {
    int t = blockIdx.x * blockDim.x + threadIdx.x;
    int total = B_ * pn * pn * 64;
    if (t >= total) return;
    int c = t & 63;
    int r = t >> 6;                 // b*pn*pn + py*pn + px
    int px = r % pn;
    int py = (r / pn) % pn;
    int b  = r / (pn * pn);
    int s = 32 / pn;
    const float* src = f_rest + (b * 64 + c) * 1024 + (py * s) * 32 + px * s;
    float sum = 0.0f;
    for (int dy = 0; dy < s; ++dy)
        for (int dx = 0; dx < s; ++dx)
            sum += src[dy * 32 + dx];
    X[t] = sum / (float)(s * s);
}

// ---------------- WMMA f32 distance GEMM + argmin, async-LDS B pipeline ----------------
// Block = 4 waves; wave w owns row tile (blockIdx.x*4 + w). All waves share the
// 256 column chunks (16 codes = contiguous 4KB of E). The next chunk's B tile is
// streamed global->LDS with GLOBAL_LOAD_ASYNC_TO_LDS_B128 (ASYNCcnt) while the
// current one feeds 16 chained V_WMMA_F32_16X16X4_F32 per wave from LDS.
__global__ __launch_bounds__(128)
void argmin_wmma_kernel(const float* __restrict__ X, const float* __restrict__ E,
                        const float* __restrict__ esq, int* __restrict__ out_idx,
                        int ntiles) {
    __shared__ float smem[2][16][64];   // double-buffered 16x64 B tile (8 KB)

    const int tid  = threadIdx.x;
    const int lane = tid & 31;
    const int wave = tid >> 5;
    const int l    = lane & 15;
    const int hi   = lane >> 4;           // half select
    const int koff = hi * 2;              // A/B: lanes 16-31 carry K=2,3
    const int tile = blockIdx.x * 4 + wave;
    const bool active = (tile < ntiles);

    // Preload A: this wave's 16x64 row block (each lane: 16 x float2)
    v2f areg[16];
    if (active) {
        const int row = tile * 16 + l;
        #pragma unroll
        for (int kk = 0; kk < 16; ++kk)
            areg[kk] = *(const v2f*)(X + row * 64 + kk * 4 + koff);
    }

    float bestv[8];
    int   besti[8];
    #pragma unroll
    for (int v = 0; v < 8; ++v) { bestv[v] = 3.4e38f; besti[v] = 0x7fffffff; }

    const unsigned lds_base = (unsigned)(uintptr_t)(void*)&smem[0][0][0];
    const unsigned o1 = (unsigned)tid * 16u;
    const unsigned o2 = 2048u + (unsigned)tid * 16u;

    // prologue: async-copy chunk 0's 4KB tile into buffer 0
    {
        unsigned long long g = (unsigned long long)(uintptr_t)(const void*)E;
        asm volatile("global_load_async_to_lds_b128 %0, %1, off"
                     :: "v"(lds_base + o1), "v"(g + o1) : "memory");
        asm volatile("global_load_async_to_lds_b128 %0, %1, off"
                     :: "v"(lds_base + o2), "v"(g + o2) : "memory");
        asm volatile("s_wait_asynccnt 0x0" ::: "memory");
    }
    __syncthreads();

    for (int chunk = 0; chunk < 256; ++chunk) {
        const int cur  = chunk & 1;
        // kick off async copy of the next tile into the other buffer
        if (chunk + 1 < 256) {
            unsigned long long g =
                (unsigned long long)(uintptr_t)(const void*)(E + (size_t)(chunk + 1) * 16 * 64);
            unsigned dst = lds_base + (unsigned)(cur ^ 1) * 4096u;
            asm volatile("global_load_async_to_lds_b128 %0, %1, off"
                         :: "v"(dst + o1), "v"(g + o1) : "memory");
            asm volatile("global_load_async_to_lds_b128 %0, %1, off"
                         :: "v"(dst + o2), "v"(g + o2) : "memory");
        }

        if (active) {
            const int col0 = chunk * 16;
            const int code = col0 + l;
            v8f c0 = {}, c1 = {};
            #pragma unroll
            for (int kk = 0; kk < 16; kk += 2) {
                v2f b0 = *(const v2f*)&smem[cur][l][kk * 4 + koff];
                v2f b1 = *(const v2f*)&smem[cur][l][kk * 4 + 4 + koff];
                c0 = __builtin_amdgcn_wmma_f32_16x16x4_f32(false, areg[kk],     false, b0,
                                                           (short)0, c0, false, false);
                c1 = __builtin_amdgcn_wmma_f32_16x16x4_f32(false, areg[kk + 1], false, b1,
                                                           (short)0, c1, false, false);
            }
            const float es = esq[code];   // column N of this lane's C/D slots
            #pragma unroll
            for (int v = 0; v < 8; ++v) {
                float val = es - 2.0f * (c0[v] + c1[v]);
                if (val < bestv[v]) { bestv[v] = val; besti[v] = code; }
            }
        }

        // drain this wave's async copies, then block-sync (read-done + write-done)
        asm volatile("s_wait_asynccnt 0x0" ::: "memory");
        __syncthreads();
    }

    if (active) {
        // cross-lane argmin within each 16-lane half (same M set per half),
        // lexicographic tie-break -> first index wins like jnp.argmin
        #pragma unroll
        for (int v = 0; v < 8; ++v) {
            for (int m = 8; m >= 1; m >>= 1) {
                float ov = __shfl_xor(bestv[v], m, 16);
                int   oi = __shfl_xor(besti[v], m, 16);
                if (ov < bestv[v] || (ov == bestv[v] && oi < besti[v])) {
                    bestv[v] = ov; besti[v] = oi;
                }
            }
        }
        if (l == 0) {
            const int base = tile * 16 + hi * 8;   // lane0 -> rows v, lane16 -> rows 8+v
            #pragma unroll
            for (int v = 0; v < 8; ++v) out_idx[base + v] = besti[v];
        }
    }
}

// ---------------- gather codes + bilinear upsample to 32x32 ----------------
__global__ void gather_up_kernel(const int* __restrict__ idx, const float* __restrict__ E,
                                 float* __restrict__ h_pre, int pn) {
    int t = blockIdx.x * blockDim.x + threadIdx.x;
    if (t >= FHAT_N) return;
    int x = t & 31, y = (t >> 5) & 31, c = (t >> 10) & 63, b = t >> 16;
    if (pn == 32) {
        int I = idx[(b * 32 + y) * 32 + x];
        h_pre[t] = E[I * 64 + c];
        return;
    }
    float scale = (float)pn / 32.0f;
    float sy = (y + 0.5f) * scale - 0.5f;
    float sx = (x + 0.5f) * scale - 0.5f;
    float fy0 = floorf(sy), fx0 = floorf(sx);
    float wy = sy - fy0, wx = sx - fx0;
    int iy0 = (int)fy0, ix0 = (int)fx0;
    int y0 = iy0 < 0 ? 0 : (iy0 > pn - 1 ? pn - 1 : iy0);
    int y1 = iy0 + 1 < 0 ? 0 : (iy0 + 1 > pn - 1 ? pn - 1 : iy0 + 1);
    int x0 = ix0 < 0 ? 0 : (ix0 > pn - 1 ? pn - 1 : ix0);
    int x1 = ix0 + 1 < 0 ? 0 : (ix0 + 1 > pn - 1 ? pn - 1 : ix0 + 1);
    const int base = b * pn * pn;
    float v00 = E[idx[base + y0 * pn + x0] * 64 + c];
    float v01 = E[idx[base + y0 * pn + x1] * 64 + c];
    float v10 = E[idx[base + y1 * pn + x0] * 64 + c];
    float v11 = E[idx[base + y1 * pn + x1] * 64 + c];
    h_pre[t] = (1.0f - wy) * ((1.0f - wx) * v00 + wx * v01)
             +          wy * ((1.0f - wx) * v10 + wx * v11);
}

// ---------------- 3x3 conv (SAME, zero pad) + 0.5/0.5 blend + state update ----------------
__global__ void conv_apply_kernel(const float* __restrict__ h_pre, const float* __restrict__ wmod,
                                  const float* __restrict__ cb, float* __restrict__ f_hat,
                                  float* __restrict__ f_rest) {
    int t = blockIdx.x * blockDim.x + threadIdx.x;
    if (t >= FHAT_N) return;
    int x = t & 31, y = (t >> 5) & 31, o = (t >> 10) & 63, b = t >> 16;
    const float* wbase = wmod + o * 64 * 9;
    const float* in_b  = h_pre + b * 65536;
    float acc = cb[o];
    for (int i = 0; i < 64; ++i) {
        const float* w9    = wbase + i * 9;
        const float* plane = in_b + i * 1024;
        #pragma unroll
        for (int ky = 0; ky < 3; ++ky) {
            int yy = y + ky - 1;
            if (yy < 0 || yy > 31) continue;
            #pragma unroll
            for (int kx = 0; kx < 3; ++kx) {
                int xx = x + kx - 1;
                if (xx < 0 || xx > 31) continue;
                acc += w9[ky * 3 + kx] * plane[yy * 32 + xx];
            }
        }
    }
    float hp = in_b[o * 1024 + y * 32 + x];
    float h  = 0.5f * hp + 0.5f * acc;     // h*(1-RESI) + y*RESI, RESI=0.5
    f_hat[t]  += h;
    f_rest[t] -= h;
}

// ---------------- vq loss: partial sums of (f_hat - f)^2 ----------------
__global__ void vq_partial_kernel(const float* __restrict__ f_hat, const float* __restrict__ f,
                                  float* __restrict__ partials) {
    __shared__ float sd[256];
    float sum = 0.0f;
    for (int i = blockIdx.x * blockDim.x + threadIdx.x; i < FHAT_N; i += gridDim.x * blockDim.x) {
        float d = f_hat[i] - f[i];
        sum += d * d;
    }
    float tot = blockReduceSum256(sum, sd);
    if (threadIdx.x == 0) partials[blockIdx.x] = tot;
}

// ---------------- commit: (down(f_hat,pn) - down(f,pn))^2 partials ----------------
__global__ void commit_partial_kernel(const float* __restrict__ f_hat, const float* __restrict__ f,
                                      float* __restrict__ partials, int pn) {
    __shared__ float sd[256];
    int s = 32 / pn;
    float inv = 1.0f / (float)(s * s);
    int M = B_ * 64 * pn * pn;
    float sum = 0.0f;
    for (int t = blockIdx.x * blockDim.x + threadIdx.x; t < M; t += gridDim.x * blockDim.x) {
        int px = t % pn;
        int py = (t / pn) % pn;
        int bc = t / (pn * pn);
        const float* a = f_hat + bc * 1024 + (py * s) * 32 + px * s;
        const float* g = f     + bc * 1024 + (py * s) * 32 + px * s;
        float sa = 0.0f, sg = 0.0f;
        for (int dy = 0; dy < s; ++dy)
            for (int dx = 0; dx < s; ++dx) {
                sa += a[dy * 32 + dx];
                sg += g[dy * 32 + dx];
            }
        float d = (sa - sg) * inv;
        sum += d * d;
    }
    float tot = blockReduceSum256(sum, sd);
    if (threadIdx.x == 0) partials[blockIdx.x] = tot;
}

// ---------------- deterministic final reduce ----------------
__global__ void reduce_final_kernel(const float* __restrict__ partials, int n, float scale,
                                    float* __restrict__ out, int accumulate) {
    __shared__ float sd[256];
    float s = 0.0f;
    for (int i = threadIdx.x; i < n; i += 256) s += partials[i];
    float tot = blockReduceSum256(s, sd);
    if (threadIdx.x == 0) {
        if (accumulate) *out += tot * scale;
        else            *out  = tot * scale;
    }
}

__global__ void write_commit_kernel(const float* __restrict__ commit_acc, float* __restrict__ out) {
    if (threadIdx.x == 0) *out = (*commit_acc) * 0.25f / 6.0f;   // *BETA / SN
}

// ---------------- host launch ----------------
extern "C" void kernel_launch(void* const* d_in, const int* in_sizes, int n_in,
                              void* d_out, int out_size, void* d_ws, size_t ws_size,
                              hipStream_t stream) {
    const float* f   = (const float*)d_in[0];
    const float* Ew  = (const float*)d_in[1];
    const float* cw  = (const float*)d_in[2];
    const float* cb  = (const float*)d_in[3];
    const float* fcw = (const float*)d_in[4];
    const float* fcb = (const float*)d_in[5];

    float* out        = (float*)d_out;
    float* f_hat      = out;               // f_hat_out == f_hat numerically (STE)
    float* vq_out     = out + FHAT_N;      // 6 slots
    float* commit_out = out + FHAT_N + 6;  // 1 slot

    char* ws = (char*)d_ws;
    float* f_rest     = (float*)ws; ws += (size_t)FHAT_N * 4;
    float* Xnc        = (float*)ws; ws += (size_t)FHAT_N * 4;   // max N*64 (pn=32)
    float* h_pre      = (float*)ws; ws += (size_t)FHAT_N * 4;
    float* esq        = (float*)ws; ws += (size_t)V_ * 4;
    float* wmod       = (float*)ws; ws += (size_t)4 * 64 * 64 * 9 * 4;
    int*   idxbuf     = (int*)ws;   ws += (size_t)32768 * 4;
    float* partials   = (float*)ws; ws += (size_t)1024 * 4;
    float* commit_acc = (float*)ws; ws += 4;

    init_kernel<<<(FHAT_N + 255) / 256, 256, 0, stream>>>(f, f_rest, f_hat, commit_acc);
    esq_kernel<<<V_ / 256, 256, 0, stream>>>(Ew, esq);
    wmod_kernel<<<(4 * 64 * 64 + 255) / 256, 256, 0, stream>>>(cw, fcw, fcb, wmod);

    const int pns[6] = {1, 2, 4, 8, 16, 32};
    const int ks[6]  = {0, 0, 1, 2, 3, 3};   // PhiPartiallyShared tick argmin, K=4, SN=6

    for (int si = 0; si < 6; ++si) {
        int pn = pns[si];
        int N  = B_ * pn * pn;
        int k  = ks[si];
        int totalX = N * 64;
        int ntiles = N / 16;

        down_nc_kernel<<<(totalX + 255) / 256, 256, 0, stream>>>(f_rest, Xnc, pn);
        argmin_wmma_kernel<<<(ntiles + 3) / 4, 128, 0, stream>>>(Xnc, Ew, esq, idxbuf, ntiles);
        gather_up_kernel<<<FHAT_N / 256, 256, 0, stream>>>(idxbuf, Ew, h_pre, pn);
        conv_apply_kernel<<<FHAT_N / 256, 256, 0, stream>>>(h_pre, wmod + k * 64 * 64 * 9,
                                                            cb + k * 64, f_hat, f_rest);

        vq_partial_kernel<<<1024, 256, 0, stream>>>(f_hat, f, partials);
        reduce_final_kernel<<<1, 256, 0, stream>>>(partials, 1024, 1.0f / (float)FHAT_N,
                                                   vq_out + si, 0);

        int M = B_ * 64 * pn * pn;
        int cgrid = (M + 255) / 256; if (cgrid > 1024) cgrid = 1024;
        commit_partial_kernel<<<cgrid, 256, 0, stream>>>(f_hat, f, partials, pn);
        reduce_final_kernel<<<1, 256, 0, stream>>>(partials, cgrid, 1.0f / (float)M,
                                                   commit_acc, 1);
    }

    write_commit_kernel<<<1, 1, 0, stream>>>(commit_acc, commit_out);
}